// QuantumStateEmbedding_60404420051235
// MI455X (gfx1250) — compile-verified
//
#include <hip/hip_runtime.h>
#include <hip/hip_bf16.h>
#include <math.h>

typedef __attribute__((ext_vector_type(2))) float v2f;
typedef __attribute__((ext_vector_type(8))) float v8f;

#define NQ 14
#define DD 16384
#define EE 2048

// ---------------------------------------------------------------------------
// Split-K GEMV via V_WMMA_F32_16X16X4_F32 (full fp32 precision).
//   y[n] = sum_k x[k] * W[k*N + n]      (W row-major, fan_out fast)
// One wave per block. A row 0 carries x[k..k+3] (lane0: K=0,1; lane16: K=2,3
// per ISA 32-bit A 16x4 layout), rows 1-15 zero. B 4x16 tile: lane L<16:
// VGPR0=W[k+0][n0+L], VGPR1=W[k+1][n0+L]; lane L>=16: K=2,3 at col n0+L-16.
// y partials land in c[0] of lanes 0..15 (C/D: VGPR0 = M0, N=lane).
// Main loop is guard-free (kChunk multiple of 4, kEnd<=K); only the final
// ragged group (K not multiple of 4) takes the guarded tail.
// ---------------------------------------------------------------------------
__global__ __launch_bounds__(32) void gemv_wmma_f32(
    const float* __restrict__ x, const float* __restrict__ W,
    float* __restrict__ part, int K, int N, int kChunk, int reluIn)
{
    const int lane = threadIdx.x;                 // 0..31, wave32
    const int n0   = blockIdx.x * 16;
    const int k0   = blockIdx.y * kChunk;
    int kEnd = k0 + kChunk; if (kEnd > K) kEnd = K;
    const int  col  = n0 + (lane & 15);
    const int  hi   = lane >> 4;                  // half-wave select
    const bool feed = ((lane & 15) == 0);         // lanes 0 and 16 carry A row 0
    const int  kBody = k0 + (((kEnd - k0) > 0 ? (kEnd - k0) : 0) & ~3);

    const float* wp = W + (size_t)(k0 + (hi << 1)) * N + col;
    const float* xp = x + k0 + (hi << 1);
    const size_t stepW = (size_t)4 * N;

    v8f c = {};                                   // zero accumulator
#pragma unroll 4
    for (int k = k0; k < kBody; k += 4) {
        v2f B;
        B.x = wp[0];
        B.y = wp[N];
        v2f A = {0.0f, 0.0f};
        if (feed) {
            float a0 = xp[0], a1 = xp[1];
            if (reluIn) { a0 = fmaxf(a0, 0.0f); a1 = fmaxf(a1, 0.0f); }
            A.x = a0; A.y = a1;
        }
        // (neg_a, A, neg_b, B, c_mod, C, reuse_a, reuse_b)
        c = __builtin_amdgcn_wmma_f32_16x16x4_f32(false, A, false, B,
                                                  (short)0, c, false, false);
        wp += stepW;
        xp += 4;
    }
    if (kBody < kEnd) {                           // ragged tail (1..3 valid k)
        const int krow = kBody + (hi << 1);
        v2f B;
        B.x = (krow     < K) ? wp[0] : 0.0f;
        B.y = (krow + 1 < K) ? wp[N] : 0.0f;
        v2f A = {0.0f, 0.0f};
        if (feed) {
            float a0 = (krow     < K) ? xp[0] : 0.0f;
            float a1 = (krow + 1 < K) ? xp[1] : 0.0f;
            if (reluIn) { a0 = fmaxf(a0, 0.0f); a1 = fmaxf(a1, 0.0f); }
            A.x = a0; A.y = a1;
        }
        c = __builtin_amdgcn_wmma_f32_16x16x4_f32(false, A, false, B,
                                                  (short)0, c, false, false);
    }
    if (lane < 16)
        part[(size_t)blockIdx.y * N + col] = c[0];
}

__global__ void reduce_bias_kernel(const float* __restrict__ part,
                                   const float* __restrict__ bias,
                                   float* __restrict__ y, int N, int gy)
{
    int n = blockIdx.x * blockDim.x + threadIdx.x;
    if (n < N) {
        float s = bias[n];
        for (int g = 0; g < gy; ++g) s += part[(size_t)g * N + n];
        y[n] = s;
    }
}

// sum(re^2 + im^2) -> sqrt, single block
__global__ void normsq_kernel(const float* __restrict__ re,
                              const float* __restrict__ im, float* outNorm)
{
    __shared__ float sm[256];
    float s = 0.f;
    for (int i = threadIdx.x; i < DD; i += 256) {
        float a = re[i], b = im[i];
        s += a * a + b * b;
    }
    sm[threadIdx.x] = s; __syncthreads();
    for (int w = 128; w > 0; w >>= 1) {
        if (threadIdx.x < w) sm[threadIdx.x] += sm[threadIdx.x + w];
        __syncthreads();
    }
    if (threadIdx.x == 0) outNorm[0] = sqrtf(sm[0]);
}

__global__ void normph_kernel(const float* __restrict__ qr,
                              const float* __restrict__ qi,
                              const float* __restrict__ nrm,
                              float* __restrict__ re_n,
                              float* __restrict__ im_n,
                              float* __restrict__ ph)
{
    int i = blockIdx.x * blockDim.x + threadIdx.x;
    if (i < DD) {
        float inv = 1.0f / nrm[0];
        float r = qr[i], m = qi[i];
        re_n[i] = r * inv;
        im_n[i] = m * inv;
        ph[i]   = atan2f(m, r);   // scale-invariant: atan2(im/n, re/n)==atan2(im,re)
    }
}

// one block per (i,j) pair, p enumerated as i:0..12, j:i+1..13
__global__ void entfeat_kernel(const float* __restrict__ re,
                               const float* __restrict__ im,
                               float* __restrict__ ef)
{
    int p = blockIdx.x;
    int i = 0, rem = p, cnt = NQ - 1;
    while (rem >= cnt) { rem -= cnt; ++i; --cnt; }
    int j  = i + 1 + rem;
    int si = 1 << (NQ - 1 - i);
    int sj = 1 << (NQ - 1 - j);
    int L  = 1 << (i + 1);

    __shared__ float srr[256], sii[256];
    float rr = 0.f, ii = 0.f;
    for (int t = threadIdx.x; t < L; t += 256) {
        float a = re[t * si], b = im[t * si];
        float c = re[1 + t * sj], d = im[1 + t * sj];
        rr += a * c - b * d;
        ii += a * d + b * c;
    }
    srr[threadIdx.x] = rr; sii[threadIdx.x] = ii; __syncthreads();
    for (int w = 128; w > 0; w >>= 1) {
        if (threadIdx.x < w) {
            srr[threadIdx.x] += srr[threadIdx.x + w];
            sii[threadIdx.x] += sii[threadIdx.x + w];
        }
        __syncthreads();
    }
    if (threadIdx.x == 0) {
        float R = srr[0], I = sii[0];
        ef[p] = sqrtf(R * R + I * I);
    }
}

__global__ void layernorm_kernel(const float* __restrict__ a,
                                 const float* __restrict__ g,
                                 const float* __restrict__ b,
                                 float* __restrict__ out, int n)
{
    __shared__ float s1[256], s2[256];
    float t1 = 0.f, t2 = 0.f;
    for (int i = threadIdx.x; i < n; i += 256) {
        float v = a[i]; t1 += v; t2 += v * v;
    }
    s1[threadIdx.x] = t1; s2[threadIdx.x] = t2; __syncthreads();
    for (int w = 128; w > 0; w >>= 1) {
        if (threadIdx.x < w) {
            s1[threadIdx.x] += s1[threadIdx.x + w];
            s2[threadIdx.x] += s2[threadIdx.x + w];
        }
        __syncthreads();
    }
    float m   = s1[0] / n;
    float var = s2[0] / n - m * m;
    float inv = rsqrtf(var + 1e-5f);
    for (int i = threadIdx.x; i < n; i += 256)
        out[i] = (a[i] - m) * inv * g[i] + b[i];
}

extern "C" void kernel_launch(void* const* d_in, const int* in_sizes, int n_in,
                              void* d_out, int out_size, void* d_ws, size_t ws_size,
                              hipStream_t stream)
{
    const float* qs_real = (const float*)d_in[0];
    const float* qs_imag = (const float*)d_in[1];
    const float* amp_w1  = (const float*)d_in[2];
    const float* amp_b1  = (const float*)d_in[3];
    const float* amp_w2  = (const float*)d_in[4];
    const float* amp_b2  = (const float*)d_in[5];
    const float* ln_g    = (const float*)d_in[6];
    const float* ln_b    = (const float*)d_in[7];
    const float* ph_w1   = (const float*)d_in[8];
    const float* ph_b1   = (const float*)d_in[9];
    const float* ph_w2   = (const float*)d_in[10];
    const float* ph_b2   = (const float*)d_in[11];
    const float* ent_w1  = (const float*)d_in[12];
    const float* ent_b1  = (const float*)d_in[13];
    const float* ent_w2  = (const float*)d_in[14];
    const float* ent_b2  = (const float*)d_in[15];
    const float* fus_w   = (const float*)d_in[16];
    const float* fus_b   = (const float*)d_in[17];
    float* out = (float*)d_out;

    // workspace layout (floats)
    float* wsf    = (float*)d_ws;
    float* nrm    = wsf;                 // 1 (padded to 16)
    float* re_n   = wsf + 16;            // D   \ contiguous -> x = re_n (len 2D)
    float* im_n   = re_n + DD;           // D   /
    float* phases = im_n + DD;           // D
    float* h_amp1 = phases + DD;         // 2048
    float* a_buf  = h_amp1 + EE;         // 2048
    float* comb   = a_buf + EE;          // 3584 = amp_emb | ph_emb | ent_emb
    float* h_ph1  = comb + 3584;         // 1024
    float* ef     = h_ph1 + 1024;        // 96
    float* h_ent1 = ef + 96;             // 512
    float* part   = h_ent1 + 512;        // up to 64*2048 floats

    auto gemv = [&](const float* x, const float* W, const float* bias, float* y,
                    int K, int N, int gy, int reluIn) {
        int kchunk = (((K + gy - 1) / gy) + 3) & ~3;
        dim3 grid(N / 16, gy);
        gemv_wmma_f32<<<grid, 32, 0, stream>>>(x, W, part, K, N, kchunk, reluIn);
        reduce_bias_kernel<<<(N + 255) / 256, 256, 0, stream>>>(part, bias, y, N, gy);
    };

    normsq_kernel<<<1, 256, 0, stream>>>(qs_real, qs_imag, nrm);
    normph_kernel<<<(DD + 255) / 256, 256, 0, stream>>>(qs_real, qs_imag, nrm,
                                                        re_n, im_n, phases);

    // amplitude branch: relu(x@W1+b1)@W2+b2, then LayerNorm
    gemv(re_n,   amp_w1, amp_b1, h_amp1, 2 * DD, EE, 64, 0);
    gemv(h_amp1, amp_w2, amp_b2, a_buf,  EE,     EE, 16, 1);
    layernorm_kernel<<<1, 256, 0, stream>>>(a_buf, ln_g, ln_b, comb, EE);

    // phase branch
    gemv(phases, ph_w1, ph_b1, h_ph1,     DD,     EE / 2, 64, 0);
    gemv(h_ph1,  ph_w2, ph_b2, comb + EE, EE / 2, EE / 2,  8, 1);

    // entanglement branch
    entfeat_kernel<<<NQ * (NQ - 1) / 2, 256, 0, stream>>>(re_n, im_n, ef);
    gemv(ef,     ent_w1, ent_b1, h_ent1,             91,     EE / 4, 1, 0);
    gemv(h_ent1, ent_w2, ent_b2, comb + EE + EE / 2, EE / 4, EE / 4, 4, 1);

    // fusion: combined (3584) @ fus_w + fus_b -> d_out (2048)
    gemv(comb, fus_w, fus_b, out, EE + EE / 2 + EE / 4, EE, 28, 0);
}